// ResonanceLayer_51891794870273
// MI455X (gfx1250) — compile-verified
//
#include <hip/hip_runtime.h>
#include <math.h>

typedef __attribute__((ext_vector_type(2))) float v2f;
typedef __attribute__((ext_vector_type(8))) float v8f;

#define N_SAMPLES 65536
#define N_OSC 16
#define N_RES 16
#define CPD 16
#define EXPR 2
#define WIN 2048
#define FRAMES 512
#define DEF_FRAMES 256
#define NCH 32  /* res*expr channels */

/* workspace layout (float offsets) */
#define WS_RES 0                                 /* 32 x 65536 */
#define WS_RC  (NCH * N_SAMPLES)                 /* 16 x 65536 */
#define WS_CONV (WS_RC + CPD * N_SAMPLES)        /* 32 x 65536 */

/* output layout: (summed, before_upsample, cs) concatenated */
#define OUT_SUMMED 0
#define OUT_BU     N_SAMPLES
#define OUT_CS     (N_SAMPLES + N_RES * FRAMES)

/* ---------------- Kernel 1: materialize + unit-norm resonances -------------- */
__global__ void __launch_bounds__(256) k_resonances(
    const float* __restrict__ damping, const float* __restrict__ mass,
    const float* __restrict__ tension, const float* __restrict__ x0_,
    const float* __restrict__ amps, float* __restrict__ ws) {
  int ch = blockIdx.x;           /* r*EXPR + e */
  int r = ch >> 1, e = ch & 1;
  int tid = threadIdx.x;
  float xd[N_OSC], om[N_OSC], ph[N_OSC], cf[N_OSC];
  for (int o = 0; o < N_OSC; ++o) {
    int idx = (o * N_RES + r) * EXPR + e;
    float m   = 1.f / (1.f + expf(-mass[idx]));
    float dmp = 30.f / (1.f + expf(-damping[idx]));
    float ten = powf(10.f, tension[idx]);
    float x0  = x0_[idx];
    float a2  = amps[idx]; a2 *= a2;
    float xdv = dmp / (2.f * m);
    float w2  = fmaxf(ten - xdv * xdv, 1e-12f);
    float omv = sqrtf(w2);
    float phv = atan2f(xdv * x0, x0 * omv);
    float av  = x0 / cosf(phv);
    xd[o] = xdv; om[o] = omv; ph[o] = phv; cf[o] = av * a2;
  }
  float* resp = ws + WS_RES + (size_t)ch * N_SAMPLES;
  float ssq = 0.f;
  const float dt = 10.f / 65535.f;
  for (int s = 0; s < N_SAMPLES / 256; ++s) {
    int k = tid + s * 256;
    float t = (float)k * dt;
    float acc = 0.f;
    for (int o = 0; o < N_OSC; ++o)
      acc += cf[o] * expf(-xd[o] * t) * cosf(om[o] * t - ph[o]);
    if (k < 10) acc *= (float)k * (1.f / 9.f);   /* ramp */
    resp[k] = acc;
    ssq += acc * acc;
  }
  __shared__ float red[256];
  red[tid] = ssq;
  __syncthreads();
  for (int off = 128; off > 0; off >>= 1) {
    if (tid < off) red[tid] += red[tid + off];
    __syncthreads();
  }
  float inv = 1.f / (sqrtf(red[0]) + 1e-8f);
  for (int s = 0; s < N_SAMPLES / 256; ++s) {
    int k = tid + s * 256;
    resp[k] *= inv;
  }
}

/* ---------------- Kernel 2a: routing matmul (before_upsample output) -------- */
__global__ void k_routed(const float* __restrict__ control,
                         const float* __restrict__ router,
                         float* __restrict__ out_bu) {
  int tid = blockIdx.x * blockDim.x + threadIdx.x;
  if (tid >= N_RES * FRAMES) return;
  int rr = tid / FRAMES, f = tid % FRAMES;
  float acc = 0.f;
  for (int c = 0; c < CPD; ++c)
    acc += control[c * FRAMES + f] * router[c * N_RES + rr];
  out_bu[rr * FRAMES + f] = acc;
}

/* ------ Kernel 2b: sparse-upsample * impulse convolution -> rc (ws) -------- */
__global__ void __launch_bounds__(256) k_impconv(
    const float* __restrict__ noise, const float* __restrict__ attack,
    const float* __restrict__ routed, float* __restrict__ ws) {
  int c = blockIdx.x;        /* channel (cpd == res pairing) */
  int seg = blockIdx.y;      /* 32 segments of 2048 samples */
  int tid = threadIdx.x;
  __shared__ float imp[WIN];
  __shared__ float rt[32];
  for (int k = tid; k < WIN; k += 256)
    imp[k] = attack[c * WIN + k] * noise[c * WIN + k];
  int fb = seg * 16 - 15;
  if (tid < 31) {
    int f = fb + tid;
    rt[tid] = (f >= 0) ? routed[c * FRAMES + f] : 0.f;
  }
  __syncthreads();
  int base = seg * 2048;
  float* rcp = ws + WS_RC + (size_t)c * N_SAMPLES;
  for (int s = 0; s < 8; ++s) {
    int n = base + s * 256 + tid;
    int f0 = n >> 7, rr = n & 127;
    float acc = 0.f;
    for (int k = 0; k < 16; ++k)
      acc += rt[f0 - k - fb] * imp[rr + (k << 7)];
    rcp[n] = acc;
  }
}

/* ------------- Kernel 3: big Toeplitz convolution via V_WMMA --------------
 * conv[n] = sum_d res[d]*rc[n-d].  Output tiles C[i][c] = conv[n0 + 16c + i].
 * Per lag-chunk d0 (K=4): A[i][j] = res[d0+i+3-j] (Toeplitz, 2 adjacent f32
 * per lane), B[j][c] = rc[n0+16c-d0-3+j] (2 adjacent f32 per lane).
 * Each wave owns TWO output tiles (n0, n0+2048) sharing every A fragment.   */
#define S3 4096
#define RCW (S3 + 4096)

#define WMMA4(Aop, Bop, Cacc) \
  __builtin_amdgcn_wmma_f32_16x16x4_f32(false, (Aop), false, (Bop), (short)0, (Cacc), false, false)

__global__ void __launch_bounds__(256) k_conv_wmma(float* __restrict__ ws) {
  int seg = blockIdx.x;            /* 16 segments of 4096 outputs */
  int ch  = blockIdx.y;            /* 32 (res,expr) channels */
  int chr = ch >> 1;               /* rc channel */
  int tid = threadIdx.x;
  int w = tid >> 5, lane = tid & 31;
  int col = lane & 15;
  bool lo = (lane < 16);
  int n0_blk = seg * 4096;
  int n0a = n0_blk + w * 256;      /* tile A outputs */
  int n0b = n0a + 2048;            /* tile B outputs */
  const float* resg = ws + WS_RES + (size_t)ch  * N_SAMPLES;
  const float* rcg  = ws + WS_RC  + (size_t)chr * N_SAMPLES;
  float* convg      = ws + WS_CONV + (size_t)ch * N_SAMPLES;

  __shared__ float lres[S3 + 32];
  __shared__ float lrc[RCW];

  v8f accA = {0.f, 0.f, 0.f, 0.f, 0.f, 0.f, 0.f, 0.f};
  v8f accB = {0.f, 0.f, 0.f, 0.f, 0.f, 0.f, 0.f, 0.f};
  int aoff = col + (lo ? 2 : 0);          /* A lane base (rel. to d0-D0) */
  int boff = (col << 4) + (lo ? 0 : 2);   /* B lane base */

  int Dlim = n0_blk + 1792 + 2048 + 253;  /* need d0 <= n0b_max + 252 */
  for (int D0 = 0; D0 < Dlim; D0 += S3) {
    for (int k = tid; k < S3 + 32; k += 256) {
      int g = D0 + k;
      lres[k] = (g < N_SAMPLES) ? resg[g] : 0.f;
    }
    int rcb = n0_blk - D0 - S3 - 2;       /* global index of lrc[0] */
    for (int k = tid; k < RCW; k += 256) {
      int g = rcb + k;
      lrc[k] = (g >= 0 && g < N_SAMPLES) ? rcg[g] : 0.f;  /* clamp kills d>n */
    }
    __syncthreads();
    int dendA = min(D0 + S3, n0a + 253);  /* wave-uniform bounds */
    int dendB = min(D0 + S3, n0b + 253);
    int d0 = D0;
    int ai = aoff;
    int biA = (n0a - n0_blk) + boff + S3 - 1;
    int biB = biA + 2048;
    /* phase 1: both tiles active, A fragment shared, unroll x8 */
    for (; d0 + 32 <= dendA; d0 += 32) {
      #pragma unroll
      for (int u = 0; u < 8; ++u) {
        int au = ai + 4 * u;
        v2f A;  A.x = lres[au + 1];  A.y = lres[au];
        int b0 = biA - 4 * u;
        v2f B0; B0.x = lrc[b0]; B0.y = lrc[b0 + 1];
        accA = WMMA4(A, B0, accA);
        int b1 = biB - 4 * u;
        v2f B1; B1.x = lrc[b1]; B1.y = lrc[b1 + 1];
        accB = WMMA4(A, B1, accB);
      }
      ai += 32; biA -= 32; biB -= 32;
    }
    for (; d0 < dendA; d0 += 4) {
      v2f A;  A.x = lres[ai + 1];  A.y = lres[ai];
      v2f B0; B0.x = lrc[biA]; B0.y = lrc[biA + 1];
      accA = WMMA4(A, B0, accA);
      v2f B1; B1.x = lrc[biB]; B1.y = lrc[biB + 1];
      accB = WMMA4(A, B1, accB);
      ai += 4; biA -= 4; biB -= 4;
    }
    /* phase 2: only far tile still accumulating */
    for (; d0 + 32 <= dendB; d0 += 32) {
      #pragma unroll
      for (int u = 0; u < 8; ++u) {
        int au = ai + 4 * u;
        v2f A;  A.x = lres[au + 1];  A.y = lres[au];
        int b1 = biB - 4 * u;
        v2f B1; B1.x = lrc[b1]; B1.y = lrc[b1 + 1];
        accB = WMMA4(A, B1, accB);
      }
      ai += 32; biB -= 32;
    }
    for (; d0 < dendB; d0 += 4) {
      v2f A;  A.x = lres[ai + 1];  A.y = lres[ai];
      v2f B1; B1.x = lrc[biB]; B1.y = lrc[biB + 1];
      accB = WMMA4(A, B1, accB);
      ai += 4; biB -= 4;
    }
    __syncthreads();
  }
  /* epilogue: missing lags d in [0, i) + store. n = n0 + 16*col + M, i = M */
  int Mbase = lo ? 0 : 8;
  for (int tile = 0; tile < 2; ++tile) {
    int n0 = tile ? n0b : n0a;
    for (int p = 0; p < 8; ++p) {
      int M = Mbase + p;
      int n = n0 + (col << 4) + M;
      float s = tile ? accB[p] : accA[p];
      for (int d = 0; d < M; ++d)
        s += resg[d] * rcg[n - d];
      convg[n] = s;
    }
  }
}

/* ---------------- Kernel 4: deform weights, mix, tanh-sum, cs -------------- */
__global__ void __launch_bounds__(256) k_final(
    const float* __restrict__ deform, const float* __restrict__ mixm,
    const float* __restrict__ gains, const float* __restrict__ ws,
    float* __restrict__ out) {
  int n = blockIdx.x * blockDim.x + threadIdx.x;
  if (n >= N_SAMPLES) return;
  const float* rc = ws + WS_RC;
  const float* cv = ws + WS_CONV;
  float cs = 0.f;
  for (int c = 0; c < CPD; ++c) cs += rc[c * N_SAMPLES + n];
  out[OUT_CS + n] = cs;
  /* linear resize of softmaxed deformation weights (half-pixel centers) */
  float u  = (n + 0.5f) * ((float)DEF_FRAMES / (float)N_SAMPLES) - 0.5f;
  float uc = fminf(fmaxf(u, 0.f), (float)(DEF_FRAMES - 1));
  int f0 = (int)uc;
  int f1 = min(f0 + 1, DEF_FRAMES - 1);
  float fr = uc - (float)f0;
  float s0 = 1.f + deform[f0], s1 = deform[DEF_FRAMES + f0];
  float mm = fmaxf(s0, s1);
  float e0 = expf(s0 - mm), e1 = expf(s1 - mm);
  float w0a = e0 / (e0 + e1);
  s0 = 1.f + deform[f1]; s1 = deform[DEF_FRAMES + f1];
  mm = fmaxf(s0, s1);
  e0 = expf(s0 - mm); e1 = expf(s1 - mm);
  float w0b = e0 / (e0 + e1);
  float dw0 = w0a + fr * (w0b - w0a);
  float dw1 = 1.f - dw0;
  float acc = 0.f;
  for (int r = 0; r < N_RES; ++r) {
    float x = dw0 * cv[(2 * r) * N_SAMPLES + n] + dw1 * cv[(2 * r + 1) * N_SAMPLES + n];
    float m0 = mixm[2 * r], m1 = mixm[2 * r + 1];
    float mx = fmaxf(m0, m1);
    float g0 = expf(m0 - mx), g1 = expf(m1 - mx);
    float wm = g0 / (g0 + g1);
    float y = wm * rc[r * N_SAMPLES + n] + (1.f - wm) * x;
    acc += tanhf(y * fabsf(gains[r]));
  }
  out[OUT_SUMMED + n] = acc;
}

extern "C" void kernel_launch(void* const* d_in, const int* in_sizes, int n_in,
                              void* d_out, int out_size, void* d_ws, size_t ws_size,
                              hipStream_t stream) {
  (void)in_sizes; (void)n_in; (void)out_size; (void)ws_size;
  const float* control = (const float*)d_in[0];
  const float* deform  = (const float*)d_in[1];
  const float* noise   = (const float*)d_in[2];
  const float* attack  = (const float*)d_in[3];
  const float* router  = (const float*)d_in[4];
  const float* mixm    = (const float*)d_in[5];
  const float* gains   = (const float*)d_in[6];
  const float* damping = (const float*)d_in[7];
  const float* mass    = (const float*)d_in[8];
  const float* tension = (const float*)d_in[9];
  const float* x0      = (const float*)d_in[10];
  const float* amps    = (const float*)d_in[11];
  float* out = (float*)d_out;
  float* ws  = (float*)d_ws;

  hipLaunchKernelGGL(k_resonances, dim3(NCH), dim3(256), 0, stream,
                     damping, mass, tension, x0, amps, ws);
  hipLaunchKernelGGL(k_routed, dim3(32), dim3(256), 0, stream,
                     control, router, out + OUT_BU);
  hipLaunchKernelGGL(k_impconv, dim3(CPD, 32), dim3(256), 0, stream,
                     noise, attack, out + OUT_BU, ws);
  hipLaunchKernelGGL(k_conv_wmma, dim3(16, 32), dim3(256), 0, stream, ws);
  hipLaunchKernelGGL(k_final, dim3(N_SAMPLES / 256), dim3(256), 0, stream,
                     deform, mixm, gains, ws, out);
}